// Model_35974646071429
// MI455X (gfx1250) — compile-verified
//
#include <hip/hip_runtime.h>

typedef float v2f __attribute__((ext_vector_type(2)));
typedef float v8f __attribute__((ext_vector_type(8)));

#define NQ    12
#define NL    6
#define DIM   4096   // 2^12 amplitudes
#define BLOCK 256    // 8 wave32s

// state flat index: wire w <-> bit (11-w). Group g occupies bits sh..sh+3, sh = 8-4g.
__host__ __device__ static constexpr int state_index(int r, int j, int sh) {
    return ((r >> sh) << (sh + 4)) | (j << sh) | (r & ((1 << sh) - 1));
}

// LDS bank swizzle (GF(2)-linear bijection): fold bits 4..7 and 8..11 into 0..3.
__host__ __device__ static constexpr int swzc(int i) {
    return i ^ ((i >> 4) & 15) ^ ((i >> 8) & 15);
}

// CNOT-ring read map (GF(2)-linear): st_new[j] = st_old[Rmap(j)].
// Ring is CNOT(w, w+1 mod 12) for w=0..11 applied in order; apply last gate's map first.
__host__ __device__ static constexpr int Rmap(int j) {
    int x = j;
    for (int w = 11; w >= 0; --w) {
        const int pc = 11 - w;
        const int pt = 11 - ((w + 1) % 12);
        x ^= ((x >> pc) & 1) << pt;
    }
    return x;
}

__global__ __launch_bounds__(BLOCK) void qsim12_kernel(
    const float* __restrict__ model_input,  // [4096, 2]
    const float* __restrict__ weights,      // [6, 12, 2]
    float* __restrict__ out)                // [4096]
{
    __shared__ float2 st2[DIM];            // interleaved (re, im), swizzled
    // Real-embedded group unitaries, K = 2*j'+comp interleaved as float2 pairs:
    //   Ar2[jo*16+ji] = ( Ur, -Ui )  -> real-part accumulator A
    //   Ai2[jo*16+ji] = ( Ui,  Ur )  -> imag-part accumulator A
    __shared__ float2 Ar2[3][256], Ai2[3][256];
    __shared__ float gates[NL][NQ][8];     // all fused 2x2 gates, precomputed
    __shared__ float red[BLOCK];

    const int b    = blockIdx.x;
    const int t    = threadIdx.x;
    const int lane = t & 31;
    const int wv   = t >> 5;

    // GF(2)-linear address bases (computed once; per-element terms are constants)
    const int storeBase = swzc(t << 4);        // swz(t*16 ^ k) = storeBase ^ k  (k<16)
    const int permBase  = swzc(Rmap(t << 4));  // swz(R(t*16 ^ k)) = permBase ^ swz(R(k))

    // |0...0> initial state (swz(0) == 0)
    for (int k = t; k < DIM; k += BLOCK) st2[k] = make_float2(0.0f, 0.0f);
    if (t == 0) st2[0] = make_float2(1.0f, 0.0f);

    const float a0 = model_input[2 * b + 0];
    const float a1 = model_input[2 * b + 1];

    // --- all per-wire fused gates G = RZ(phi)*RY(theta_total), hoisted out of layers ---
    if (t < NL * NQ) {
        const int l = t / NQ, w = t % NQ;
        const int limit = NQ - (l >> 1);
        float th = weights[(l * NQ + w) * 2 + 0];
        if (w < limit) th += (w & 1) ? a1 : a0;      // data re-upload RY folds in
        const float ph = weights[(l * NQ + w) * 2 + 1];
        float c, s, pr, pi;
        __sincosf(0.5f * th, &s, &c);
        __sincosf(0.5f * ph, &pi, &pr);
        gates[l][w][0] =  pr * c;  gates[l][w][1] = -pi * c;   // g00 = e^{-i phi/2} c
        gates[l][w][2] = -pr * s;  gates[l][w][3] =  pi * s;   // g01 = -e^{-i phi/2} s
        gates[l][w][4] =  pr * s;  gates[l][w][5] =  pi * s;   // g10 = e^{+i phi/2} s
        gates[l][w][6] =  pr * c;  gates[l][w][7] =  pi * c;   // g11 = e^{+i phi/2} c
    }
    __syncthreads();

    #pragma unroll 1
    for (int l = 0; l < NL; ++l) {
        // --- build all three 16x16 group unitaries (tensor product of four 2x2 gates) ---
        {
            const int jo = t >> 4, ji = t & 15;
            #pragma unroll
            for (int g = 0; g < 3; ++g) {
                float re = 1.0f, im = 0.0f;
                #pragma unroll
                for (int k = 0; k < 4; ++k) {
                    const int w  = 4 * g + k;
                    const int bo = (jo >> (3 - k)) & 1;
                    const int bi = (ji >> (3 - k)) & 1;
                    const float er = gates[l][w][(bo * 2 + bi) * 2 + 0];
                    const float ei = gates[l][w][(bo * 2 + bi) * 2 + 1];
                    const float nr = re * er - im * ei;
                    const float ni = re * ei + im * er;
                    re = nr; im = ni;
                }
                Ar2[g][t] = make_float2(re, -im);
                Ai2[g][t] = make_float2(im,  re);
            }
        }
        __syncthreads();

        // --- three 4-wire groups: 16x16 complex unitary via f32 WMMA (real K=32 embedding) ---
        #pragma unroll
        for (int g = 0; g < 3; ++g) {
            const int sh = 8 - 4 * g;

            // A fragments (16x4 f32): lanes 0-15 = M; vgpr{0,1} = K{0,1} (lo) / K{2,3} (hi).
            // Global K = 4c + 2*hi + {0,1} = one float2 pair at column index 2c + hi.
            const int m  = lane & 15;
            const int hi = lane >> 4;
            v2f ar[8], ai[8];
            #pragma unroll
            for (int c = 0; c < 8; ++c) {
                const int col = m * 16 + 2 * c + hi;
                ar[c] = *(const v2f*)&Ar2[g][col];
                ai[c] = *(const v2f*)&Ai2[g][col];
            }

            // each wave owns 2 disjoint 16-column tiles of r -> no intra-group barriers
            #pragma unroll
            for (int tt = 0; tt < 2; ++tt) {
                const int r0 = (wv * 2 + tt) * 16;
                const int rn = r0 + (lane & 15);
                // Linearized swizzled addressing: swz(ri ^ (jp<<sh)) = swz(ri) ^ swz(jp<<sh)
                const int sri = swzc(state_index(rn, 0, sh));
                const int sbB = sri ^ (hi ? swzc(1 << sh) : 0);   // B: jp = 2c + hi
                const int sbD = sri ^ (hi ? swzc(8 << sh) : 0);   // D: j  = v + 8*hi

                // B fragment chunk c: lane holds (re, im) of state element j' = 2c + hi
                // at K = 4c+2hi, 4c+2hi+1 -> exactly one b64 load, no repacking.
                v2f bb[8];
                #pragma unroll
                for (int c = 0; c < 8; ++c) {
                    bb[c] = *(const v2f*)&st2[sbB ^ swzc((2 * c) << sh)];
                }
                v8f dr = {0, 0, 0, 0, 0, 0, 0, 0};
                v8f di = {0, 0, 0, 0, 0, 0, 0, 0};
                #pragma unroll
                for (int c = 0; c < 8; ++c) {
                    dr = __builtin_amdgcn_wmma_f32_16x16x4_f32(false, ar[c], false, bb[c], (short)0, dr, false, false);
                    di = __builtin_amdgcn_wmma_f32_16x16x4_f32(false, ai[c], false, bb[c], (short)0, di, false, false);
                }
                // D layout: vgpr v -> row v (lanes 0-15) / v+8 (lanes 16-31), col = lane&15
                #pragma unroll
                for (int v = 0; v < 8; ++v) {
                    st2[sbD ^ swzc(v << sh)] = make_float2(dr[v], di[v]);
                }
            }
            __syncthreads();
        }

        // --- CNOT ring: fused GF(2)-linear permutation, linearized addressing ---
        float2 tmp[16];
        #pragma unroll
        for (int k = 0; k < 16; ++k) {
            tmp[k] = st2[permBase ^ swzc(Rmap(k))];   // swz(R(k)) folds to a constant
        }
        __syncthreads();
        #pragma unroll
        for (int k = 0; k < 16; ++k) {
            st2[storeBase ^ k] = tmp[k];              // swz(k) == k for k < 16
        }
        __syncthreads();
    }

    // --- <Z> on wire 11 (logical bit 0): sum (-1)^{i&1} |a_i|^2 ---
    float acc = 0.0f;
    #pragma unroll
    for (int k = 0; k < 16; ++k) {
        const float2 v = st2[storeBase ^ k];
        const float p = v.x * v.x + v.y * v.y;
        acc += (k & 1) ? -p : p;                      // logical i = t*16+k, sign = k&1
    }
    red[t] = acc;
    __syncthreads();
    #pragma unroll
    for (int off = BLOCK / 2; off > 0; off >>= 1) {
        if (t < off) red[t] += red[t + off];
        __syncthreads();
    }
    if (t == 0) out[b] = red[0];
}

extern "C" void kernel_launch(void* const* d_in, const int* in_sizes, int n_in,
                              void* d_out, int out_size, void* d_ws, size_t ws_size,
                              hipStream_t stream) {
    const float* model_input = (const float*)d_in[0];  // [4096,2] f32
    const float* weights     = (const float*)d_in[1];  // [6,12,2] f32
    float* out = (float*)d_out;                        // [4096]  f32
    const int batch = in_sizes[0] / 2;                 // 4096
    qsim12_kernel<<<batch, BLOCK, 0, stream>>>(model_input, weights, out);
}